// MessagePassingLayer_2534030704715
// MI455X (gfx1250) — compile-verified
//
#include <hip/hip_runtime.h>
#include <hip/hip_bf16.h>

typedef __attribute__((ext_vector_type(2))) float v2f;
typedef __attribute__((ext_vector_type(8))) float v8f;

#define IN_DIM  128
#define OUT_DIM 128
#define K_TOT   256   // IN_DIM + OUT_DIM

// ---------------------------------------------------------------------------
// Packed position of W[o][k] inside the WMMA-B-fragment-ordered buffer.
// B-frag layout (16x16x4 f32): lane = half*16 + (o&15), VGPR v holds K = k0+2*half+v.
// ---------------------------------------------------------------------------
__device__ __forceinline__ size_t wpack_pos(int o, int k) {
    int ot = o >> 4, nn = o & 15;
    int k4 = k >> 2, kk = k & 3, half = kk >> 1, v = kk & 1;
    int lane = half * 16 + nn;
    return (((size_t)ot * 64 + (size_t)k4) * 32 + (size_t)lane) * 2 + (size_t)v;
}

// ---------------------------------------------------------------------------
// Utility zero kernels
// ---------------------------------------------------------------------------
__global__ void zero_f4_kernel(float4* __restrict__ p, int n4) {
    int i = blockIdx.x * blockDim.x + threadIdx.x;
    if (i < n4) p[i] = make_float4(0.f, 0.f, 0.f, 0.f);
}
__global__ void zero_i32_kernel(int* __restrict__ p, int n) {
    int i = blockIdx.x * blockDim.x + threadIdx.x;
    if (i < n) p[i] = 0;
}

// ---------------------------------------------------------------------------
// Weight prep: Wfold = Wupd[:,128:] @ Wmsg, pack [W1 | Wfold] in B-frag order.
// ---------------------------------------------------------------------------
__global__ void prep_weights_kernel(const float* __restrict__ W_msg,  // [128,128]
                                    const float* __restrict__ W_upd,  // [128,256]
                                    float* __restrict__ Wpack) {
    int o = blockIdx.x;
    int i = threadIdx.x;
    const float* wu2 = W_upd + (size_t)o * K_TOT + IN_DIM;
    float acc = 0.f;
#pragma unroll 8
    for (int j = 0; j < OUT_DIM; ++j)
        acc = fmaf(wu2[j], W_msg[(size_t)j * IN_DIM + i], acc);
    Wpack[wpack_pos(o, i)]          = W_upd[(size_t)o * K_TOT + i];
    Wpack[wpack_pos(o, i + IN_DIM)] = acc;
}

// ===========================================================================
// CSR path: histogram -> scan -> bucket fill -> gather aggregation (no fp atomics)
// ===========================================================================
__global__ void histogram_kernel(const int* __restrict__ edge_index, // [2,E]
                                 int* __restrict__ counts, int E) {
    int i = blockIdx.x * blockDim.x + threadIdx.x;
    int stride = gridDim.x * blockDim.x;
    for (int e = i; e < E; e += stride)
        atomicAdd(&counts[edge_index[E + e]], 1);
}

// Single-workgroup chunked exclusive scan (N ~ 50k -> ~49 chunks of 1024).
__global__ void __launch_bounds__(1024)
scan_kernel(const int* __restrict__ counts, int* __restrict__ offsets, int N) {
    __shared__ int temp[1024];
    __shared__ int s_carry;
    if (threadIdx.x == 0) s_carry = 0;
    __syncthreads();
    for (int base = 0; base < N; base += 1024) {
        int i = base + (int)threadIdx.x;
        int v = (i < N) ? counts[i] : 0;
        temp[threadIdx.x] = v;
        __syncthreads();
        for (int off = 1; off < 1024; off <<= 1) {
            int add = (threadIdx.x >= (unsigned)off) ? temp[threadIdx.x - off] : 0;
            __syncthreads();
            temp[threadIdx.x] += add;
            __syncthreads();
        }
        int carry = s_carry;
        int incl  = temp[threadIdx.x];
        if (i < N) offsets[i] = carry + incl - v;   // exclusive
        __syncthreads();
        if (threadIdx.x == 0) s_carry = carry + temp[1023];
        __syncthreads();
    }
    if (threadIdx.x == 0) offsets[N] = s_carry;
}

__global__ void fill_kernel(const int* __restrict__ edge_index, // [2,E]
                            int* __restrict__ cursor,
                            int* __restrict__ edge_src, int E) {
    int i = blockIdx.x * blockDim.x + threadIdx.x;
    int stride = gridDim.x * blockDim.x;
    for (int e = i; e < E; e += stride) {
        int dst = edge_index[E + e];
        int pos = atomicAdd(&cursor[dst], 1);
        edge_src[pos] = edge_index[e];
    }
}

// One wave per node: sum h[src] over the node's in-edges. h is L2-resident.
__global__ void aggregate_kernel(const float* __restrict__ h,
                                 const int* __restrict__ offsets,
                                 const int* __restrict__ edge_src,
                                 float* __restrict__ agg, int N) {
    int lane = threadIdx.x & 31;
    int w    = (blockIdx.x * blockDim.x + threadIdx.x) >> 5;
    if (w >= N) return;
    int jb = offsets[w];
    int je = offsets[w + 1];
    float4 acc = make_float4(0.f, 0.f, 0.f, 0.f);
    for (int j = jb; j < je; ++j) {
        int src = edge_src[j];
        const float4 v = *(const float4*)(h + (size_t)src * IN_DIM + lane * 4);
        acc.x += v.x; acc.y += v.y; acc.z += v.z; acc.w += v.w;
    }
    *(float4*)(agg + (size_t)w * IN_DIM + lane * 4) = acc;
}

// ===========================================================================
// Fallback path: direct fp32 atomic scatter (used if workspace is small)
// ===========================================================================
__global__ void scatter_edges_kernel(const float* __restrict__ h,
                                     const int* __restrict__ edge_index,
                                     float* __restrict__ agg, int E) {
    int lane = threadIdx.x & 31;
    int wid  = (blockIdx.x * blockDim.x + threadIdx.x) >> 5;
    int nw   = (gridDim.x * blockDim.x) >> 5;
    for (int e = wid; e < E; e += nw) {
        int src = edge_index[e];
        int dst = edge_index[E + e];
        const float4 v = *(const float4*)(h + (size_t)src * IN_DIM + lane * 4);
        float* p = agg + (size_t)dst * IN_DIM + lane * 4;
        unsafeAtomicAdd(p + 0, v.x);
        unsafeAtomicAdd(p + 1, v.y);
        unsafeAtomicAdd(p + 2, v.z);
        unsafeAtomicAdd(p + 3, v.w);
    }
}

// ===========================================================================
// Fused node GEMM + bias + ReLU via V_WMMA_F32_16X16X4_F32.
// out = relu([h | agg] @ Wpack^T + b); agg may alias out (A-tile staged to LDS
// before any store; each block only touches its own 16 rows).
// ===========================================================================
#define LDSA_STRIDE 260   // 260 % 64 == 4 -> conflict-free fragment reads

__global__ void __launch_bounds__(256)
node_update_kernel(const float* __restrict__ h,
                   const float* __restrict__ agg,
                   const float* __restrict__ Wpack,
                   const float* __restrict__ bias,
                   float* __restrict__ out, int N) {
    __shared__ float ldsA[16 * LDSA_STRIDE];

    const int n0 = blockIdx.x * 16;
    const int t  = threadIdx.x;

    // ---- Stage A2 = [h | agg] rows n0..n0+15 (16 x 256 f32) into LDS ----
    {
        int r     = t >> 4;
        int cbase = (t & 15) * 16;
        int row   = n0 + r; if (row >= N) row = N - 1;
#pragma unroll
        for (int q = 0; q < 4; ++q) {
            int c = cbase + q * 4;
            const float* ph = h   + (size_t)row * IN_DIM + c;
            const float* pa = agg + (size_t)row * IN_DIM + (c - IN_DIM);
            const float* p  = (c < IN_DIM) ? ph : pa;   // pointer select, no divergence
            float4 v = *(const float4*)p;
            *(float4*)&ldsA[r * LDSA_STRIDE + c] = v;
        }
    }
    __syncthreads();

    const int wave = t >> 5;
    const int lane = t & 31;
    const int m    = lane & 15;
    const int hh   = lane >> 4;

    v8f acc = {};
    const float2* bptr = (const float2*)(Wpack) + (size_t)wave * 64 * 32 + lane;
#pragma unroll 4
    for (int k0 = 0; k0 < K_TOT; k0 += 4) {
        const float2 av = *(const float2*)&ldsA[m * LDSA_STRIDE + k0 + 2 * hh];
        v2f a; a.x = av.x; a.y = av.y;
        const float2 bv = bptr[(k0 >> 2) * 32];
        v2f b; b.x = bv.x; b.y = bv.y;
        acc = __builtin_amdgcn_wmma_f32_16x16x4_f32(
            false, a, false, b, (short)0, acc, false, false);
    }

    // ---- Epilogue: bias + ReLU. D layout: VGPR r -> row r + 8*hh, lane&15 -> col.
    const int   o  = wave * 16 + m;
    const float bb = bias[o];
    float* obase = out + (size_t)(n0 + 8 * hh) * OUT_DIM + o;
    if (n0 + 16 <= N) {          // block-uniform fast path: no EXEC churn
#pragma unroll
        for (int r = 0; r < 8; ++r)
            obase[(size_t)r * OUT_DIM] = fmaxf(acc[r] + bb, 0.f);
    } else {
#pragma unroll
        for (int r = 0; r < 8; ++r) {
            int node = n0 + r + 8 * hh;
            if (node < N) out[(size_t)node * OUT_DIM + o] = fmaxf(acc[r] + bb, 0.f);
        }
    }
}

// ---------------------------------------------------------------------------
// Host launcher
// ---------------------------------------------------------------------------
extern "C" void kernel_launch(void* const* d_in, const int* in_sizes, int n_in,
                              void* d_out, int out_size, void* d_ws, size_t ws_size,
                              hipStream_t stream) {
    const float* h     = (const float*)d_in[0];   // [N,128]
    const int*   eidx  = (const int*)d_in[1];     // [2,E]
    const float* W_msg = (const float*)d_in[2];   // [128,128]
    const float* W_upd = (const float*)d_in[3];   // [128,256]
    const float* b_upd = (const float*)d_in[4];   // [128]
    float*       out   = (float*)d_out;           // [N,128]

    const int N = in_sizes[0] / IN_DIM;
    const int E = in_sizes[1] / 2;

    char* ws = (char*)d_ws;
    float* Wpack = (float*)ws;                      // 128 KB
    size_t wpackBytes = (size_t)OUT_DIM * K_TOT * 4;

    // CSR workspace layout (after Wpack)
    size_t offCounts  = wpackBytes;
    size_t offOffsets = offCounts  + (size_t)N * 4;
    size_t offCursor  = offOffsets + (size_t)(N + 1) * 4;
    size_t offEsrc    = offCursor  + (size_t)N * 4;
    size_t neededCSR  = offEsrc    + (size_t)E * 4;

    float* agg = out;   // reuse d_out as raw-aggregation buffer

    // Weight fold+pack (independent of graph phase)
    prep_weights_kernel<<<OUT_DIM, IN_DIM, 0, stream>>>(W_msg, W_upd, Wpack);

    if (ws_size >= neededCSR) {
        // ---- CSR path: no fp atomics ----
        int* counts   = (int*)(ws + offCounts);
        int* offsets  = (int*)(ws + offOffsets);
        int* cursor   = (int*)(ws + offCursor);
        int* edge_src = (int*)(ws + offEsrc);

        zero_i32_kernel<<<(N + 255) / 256, 256, 0, stream>>>(counts, N);
        histogram_kernel<<<(E + 255) / 256, 256, 0, stream>>>(eidx, counts, E);
        scan_kernel<<<1, 1024, 0, stream>>>(counts, offsets, N);
        hipMemcpyAsync(cursor, offsets, (size_t)N * 4, hipMemcpyDeviceToDevice, stream);
        fill_kernel<<<(E + 255) / 256, 256, 0, stream>>>(eidx, cursor, edge_src, E);
        aggregate_kernel<<<(N * 32 + 255) / 256, 256, 0, stream>>>(h, offsets, edge_src, agg, N);
    } else {
        // ---- Fallback: fp32 atomic scatter ----
        int n4 = (N * IN_DIM) / 4;
        zero_f4_kernel<<<(n4 + 255) / 256, 256, 0, stream>>>((float4*)agg, n4);
        scatter_edges_kernel<<<16384, 256, 0, stream>>>(h, eidx, agg, E);
    }

    // ---- Fused node GEMM + bias + ReLU (WMMA f32 16x16x4) ----
    node_update_kernel<<<(N + 15) / 16, 256, 0, stream>>>(h, agg, Wpack, b_upd, out, N);
}